// Traj_Model_91336774517438
// MI455X (gfx1250) — compile-verified
//
#include <hip/hip_runtime.h>
#include <cstdint>

static constexpr int NT = 16384;   // B*T tokens
static constexpr int Dm = 1024;    // model dim
static constexpr int NE = 8;       // experts
static constexpr int F4 = 4096;    // expert hidden dim
static constexpr int BM = 32;      // token rows per block
static constexpr int FC = 128;     // F4 chunk per iteration

typedef float v2f __attribute__((ext_vector_type(2)));
typedef float v8f __attribute__((ext_vector_type(8)));
typedef int   si4 __attribute__((ext_vector_type(4)));
typedef int   si8 __attribute__((ext_vector_type(8)));
typedef unsigned su4 __attribute__((ext_vector_type(4)));

__device__ __forceinline__ float gelu_tanh(float x) {
    float t = 0.7978845608028654f * (x + 0.044715f * x * x * x);
    return 0.5f * x * (1.0f + tanhf(t));
}

// TDM gather: DMA up to 16 rows of 1024 f32 (4KB each) from x into LDS.
// D# gather mode, 16-bit row indices, data_size=4B. Uniform (SGPR) inputs.
__device__ __forceinline__ void tdm_gather16(unsigned lds_off, const float* gbase,
                                             const unsigned* tk, int nv) {
    const unsigned long long ga = (unsigned long long)(uintptr_t)gbase;
    su4 g0;
    g0.x = 0x80000001u;                               // count=1 | gather_mode | 16-bit idx
    g0.y = lds_off;                                   // LDS byte offset
    g0.z = (unsigned)(ga & 0xffffffffull);            // global_addr[31:0]
    g0.w = (unsigned)((ga >> 32) & 0x01ffffffull)     // global_addr[56:32]
         | 0x80000000u;                               // type=2 ("image")
    si8 g1;
    g1[0] = 0x00020000;                               // wg_mask=0 | data_size=2 (4B)
    g1[1] = (int)(1024u << 16);                       // tensor_dim0[15:0]=1024
    g1[2] = (int)(16384u << 16);                      // dim0 hi=0 | tensor_dim1 lo=NT
    g1[3] = (int)(1024u << 16);                       // dim1 hi=0 | tile_dim0=1024
    g1[4] = nv;                                       // tile_dim1 = #valid indices
    g1[5] = 1024;                                     // tensor_dim0_stride
    g1[6] = 0;
    g1[7] = 0;
    si4 g2, g3;
    g2.x = (int)(tk[0]  | (tk[1]  << 16));
    g2.y = (int)(tk[2]  | (tk[3]  << 16));
    g2.z = (int)(tk[4]  | (tk[5]  << 16));
    g2.w = (int)(tk[6]  | (tk[7]  << 16));
    g3.x = (int)(tk[8]  | (tk[9]  << 16));
    g3.y = (int)(tk[10] | (tk[11] << 16));
    g3.z = (int)(tk[12] | (tk[13] << 16));
    g3.w = (int)(tk[14] | (tk[15] << 16));
    asm volatile("tensor_load_to_lds %0, %1, %2, %3"
                 :: "s"(g0), "s"(g1), "s"(g2), "s"(g3)
                 : "memory");
}

// ---------------- Router: one wave32 per token ----------------
__global__ __launch_bounds__(256) void router_kernel(
    const float* __restrict__ x, const float* __restrict__ noise,
    const float* __restrict__ wr, const float* __restrict__ br,
    const float* __restrict__ wn, const float* __restrict__ bn,
    float* __restrict__ gate1, float* __restrict__ gates,
    int* __restrict__ counts, int* __restrict__ lists)
{
    const int tok  = (int)((blockIdx.x * blockDim.x + threadIdx.x) >> 5);
    const int lane = (int)(threadIdx.x & 31u);
    if (tok >= NT) return;

    const float* xr = x + (size_t)tok * Dm;
    float aR[NE], aN[NE];
#pragma unroll
    for (int e = 0; e < NE; ++e) { aR[e] = 0.f; aN[e] = 0.f; }

    for (int i = 0; i < Dm / 32; ++i) {
        const int d = i * 32 + lane;
        const float xv = xr[d];
        const float* wrp = wr + (size_t)d * NE;
        const float* wnp = wn + (size_t)d * NE;
#pragma unroll
        for (int e = 0; e < NE; ++e) {
            aR[e] = fmaf(xv, wrp[e], aR[e]);
            aN[e] = fmaf(xv, wnp[e], aN[e]);
        }
    }
#pragma unroll
    for (int off = 16; off > 0; off >>= 1) {
#pragma unroll
        for (int e = 0; e < NE; ++e) {
            aR[e] += __shfl_xor(aR[e], off, 32);
            aN[e] += __shfl_xor(aN[e], off, 32);
        }
    }
    if (lane == 0) {
        float ny[NE];
#pragma unroll
        for (int e = 0; e < NE; ++e) {
            const float z  = aN[e] + bn[e];
            const float sp = fmaxf(z, 0.f) + log1pf(expf(-fabsf(z)));  // softplus
            ny[e] = aR[e] + br[e] + noise[(size_t)tok * NE + e] * sp;
        }
        float m = ny[0];
#pragma unroll
        for (int e = 1; e < NE; ++e) m = fmaxf(m, ny[e]);
        float p[NE], s = 0.f;
#pragma unroll
        for (int e = 0; e < NE; ++e) { p[e] = expf(ny[e] - m); s += p[e]; }
        const float inv_s = 1.f / s;
#pragma unroll
        for (int e = 0; e < NE; ++e) gate1[(size_t)tok * NE + e] = p[e] * inv_s;

        int i1 = 0;
#pragma unroll
        for (int e = 1; e < NE; ++e) if (ny[e] > ny[i1]) i1 = e;
        int i2 = (i1 == 0) ? 1 : 0;
#pragma unroll
        for (int e = 0; e < NE; ++e) if (e != i1 && ny[e] > ny[i2]) i2 = e;

        const float p2  = expf(ny[i2] - ny[i1]);
        const float inv = 1.f / (1.f + p2);
#pragma unroll
        for (int e = 0; e < NE; ++e) gates[(size_t)tok * NE + e] = 0.f;
        gates[(size_t)tok * NE + i1] = inv;
        gates[(size_t)tok * NE + i2] = p2 * inv;

        const int q1 = atomicAdd(&counts[i1], 1);
        lists[(size_t)i1 * NT + q1] = tok;
        const int q2 = atomicAdd(&counts[i2], 1);
        lists[(size_t)i2 * NT + q2] = tok;
    }
}

// ------------- Fused expert FFN: gelu(X@W1+b1)@W2+b2 -------------
__global__ __launch_bounds__(256) void moe_ffn_kernel(
    const float* __restrict__ x,
    const float* __restrict__ w1, const float* __restrict__ b1,
    const float* __restrict__ w2, const float* __restrict__ b2,
    const float* __restrict__ gates,
    const int* __restrict__ lists, const int* __restrict__ counts,
    float* __restrict__ out, int expert)
{
    extern __shared__ float smem[];
    float* Xs = smem;                          // [BM][Dm]  (128 KB, LDS offset 0)
    float* Hs = smem + (size_t)BM * Dm;        // [BM][FC]  (16 KB)
    int*   toks  = (int*)(Hs + (size_t)BM * FC);
    float* gvals = (float*)(toks + BM);

    const int cnt   = counts[expert];
    const int tile0 = (int)blockIdx.x * BM;
    if (tile0 >= cnt) return;

    const int tid    = (int)threadIdx.x;
    const int lane   = tid & 31;
    const int wv     = tid >> 5;            // wave 0..7
    const int m16    = lane & 15;
    const int kb     = (lane >> 4) * 2;
    const int rowoff = (lane >> 4) * 8;

    int nv = cnt - tile0; if (nv > BM) nv = BM;   // valid rows in this tile

    if (tid < BM) {
        const int r = tile0 + tid;
        const int t = (r < cnt) ? lists[(size_t)expert * NT + r] : -1;
        toks[tid]  = t;
        gvals[tid] = (t >= 0) ? gates[(size_t)t * NE + expert] : 0.f;
    }

    // Zero-fill pad rows (disjoint from the TDM-written rows [0, nv)).
    {
        const float4 z4 = make_float4(0.f, 0.f, 0.f, 0.f);
        float4* pad = (float4*)(Xs + (size_t)nv * Dm);
        const int n4 = (BM - nv) * (Dm / 4);
        for (int i = tid; i < n4; i += 256) pad[i] = z4;
    }

    // Wave 0: DMA-gather the routed token rows into LDS via the TDM.
    if (wv == 0) {
        const size_t base = (size_t)expert * NT + tile0;
        unsigned tk[BM];
#pragma unroll
        for (int i = 0; i < BM; ++i)
            tk[i] = (tile0 + i < cnt) ? (unsigned)lists[base + i] : 0u;
        const int nv0 = (nv < 16) ? nv : 16;
        tdm_gather16(0u, x, tk, nv0);
        if (nv > 16) tdm_gather16((unsigned)(16 * Dm * sizeof(float)), x, tk + 16, nv - 16);
        __builtin_amdgcn_s_wait_tensorcnt(0);
    }
    __syncthreads();

    // Wave wv owns output cols [wv*128, wv*128+128), rows 0..31 -> 16 acc tiles
    v8f acc[16] = {};

    for (int fc = 0; fc < F4; fc += FC) {
        // ---- GEMM1: h[0:32, fc+wv*16 : +16] = X @ W1 slice, K = Dm ----
        v8f h0 = {}, h1 = {};
        {
            const float* w1p = w1 + (size_t)fc + (size_t)wv * 16 + m16;
            const float* xs0 = Xs + (size_t)m16 * Dm + kb;
            const float* xs1 = Xs + (size_t)(16 + m16) * Dm + kb;
            for (int k = 0; k < Dm; k += 4) {
                const v2f a0 = *(const v2f*)(xs0 + k);
                const v2f a1 = *(const v2f*)(xs1 + k);
                v2f b;
                b.x = w1p[(size_t)(k + kb) * F4];
                b.y = w1p[(size_t)(k + kb + 1) * F4];
                h0 = __builtin_amdgcn_wmma_f32_16x16x4_f32(false, a0, false, b, (short)0, h0, false, false);
                h1 = __builtin_amdgcn_wmma_f32_16x16x4_f32(false, a1, false, b, (short)0, h1, false, false);
            }
        }
        const float bia = b1[fc + wv * 16 + m16];
#pragma unroll
        for (int j = 0; j < 8; ++j) {
            Hs[(size_t)(rowoff + j) * FC + wv * 16 + m16]      = gelu_tanh(h0[j] + bia);
            Hs[(size_t)(16 + rowoff + j) * FC + wv * 16 + m16] = gelu_tanh(h1[j] + bia);
        }
        __syncthreads();

        // ---- GEMM2: acc += h_chunk @ W2[fc:fc+FC, :] ----
        {
            const float* hs0 = Hs + (size_t)m16 * FC + kb;
            const float* hs1 = Hs + (size_t)(16 + m16) * FC + kb;
            const float* w2p = w2 + (size_t)fc * Dm + (size_t)(wv * 8) * 16 + m16;
            for (int k = 0; k < FC; k += 4) {
                const v2f a0 = *(const v2f*)(hs0 + k);
                const v2f a1 = *(const v2f*)(hs1 + k);
                const float* wk0 = w2p + (size_t)(k + kb) * Dm;
                const float* wk1 = wk0 + Dm;
#pragma unroll
                for (int j = 0; j < 8; ++j) {
                    v2f b;
                    b.x = wk0[j * 16];
                    b.y = wk1[j * 16];
                    acc[j]     = __builtin_amdgcn_wmma_f32_16x16x4_f32(false, a0, false, b, (short)0, acc[j], false, false);
                    acc[8 + j] = __builtin_amdgcn_wmma_f32_16x16x4_f32(false, a1, false, b, (short)0, acc[8 + j], false, false);
                }
            }
        }
        __syncthreads();
    }

    // Epilogue: out[token] += gate * (acc + b2). Race-free per expert launch.
#pragma unroll
    for (int jt = 0; jt < 8; ++jt) {
        const int col  = (wv * 8 + jt) * 16 + m16;
        const float bv = b2[col];
#pragma unroll
        for (int j = 0; j < 8; ++j) {
            {
                const int row = rowoff + j;
                const int t = toks[row];
                if (t >= 0) {
                    float* p = out + (size_t)t * Dm + col;
                    *p = *p + gvals[row] * (acc[jt][j] + bv);
                }
            }
            {
                const int row = 16 + rowoff + j;
                const int t = toks[row];
                if (t >= 0) {
                    float* p = out + (size_t)t * Dm + col;
                    *p = *p + gvals[row] * (acc[8 + jt][j] + bv);
                }
            }
        }
    }
}

extern "C" void kernel_launch(void* const* d_in, const int* in_sizes, int n_in,
                              void* d_out, int out_size, void* d_ws, size_t ws_size,
                              hipStream_t stream) {
    (void)in_sizes; (void)n_in; (void)out_size; (void)ws_size;
    const float* x     = (const float*)d_in[0];
    const float* noise = (const float*)d_in[1];
    const float* wr    = (const float*)d_in[2];
    const float* br    = (const float*)d_in[3];
    const float* wn    = (const float*)d_in[4];
    const float* bn    = (const float*)d_in[5];
    const float* w1    = (const float*)d_in[6];
    const float* b1    = (const float*)d_in[7];
    const float* w2    = (const float*)d_in[8];
    const float* b2    = (const float*)d_in[9];

    float* out   = (float*)d_out;             // [NT, Dm]
    float* gate1 = out + (size_t)NT * Dm;     // [NT, NE]

    float* gates  = (float*)d_ws;                                    // NT*NE f32
    int*   counts = (int*)((char*)d_ws + (size_t)NT * NE * 4);       // NE i32
    int*   lists  = (int*)((char*)d_ws + (size_t)NT * NE * 4 + 256); // NE*NT i32

    hipMemsetAsync(out, 0, (size_t)NT * Dm * sizeof(float), stream);
    hipMemsetAsync(counts, 0, NE * sizeof(int), stream);

    router_kernel<<<NT / 8, 256, 0, stream>>>(x, noise, wr, br, wn, bn,
                                              gate1, gates, counts, lists);

    const size_t smem = (size_t)(BM * Dm + BM * FC) * sizeof(float)
                      + (size_t)BM * (sizeof(int) + sizeof(float));
    const int nblk = NT / BM;
    for (int e = 0; e < NE; ++e) {
        moe_ffn_kernel<<<nblk, 256, smem, stream>>>(
            x,
            w1 + (size_t)e * Dm * F4, b1 + (size_t)e * F4,
            w2 + (size_t)e * F4 * Dm, b2 + (size_t)e * Dm,
            gates, lists, counts, out, e);
    }
}